// GraphConvolution_22660247453734
// MI455X (gfx1250) — compile-verified
//
#include <hip/hip_runtime.h>

typedef float v2f __attribute__((ext_vector_type(2)));
typedef float v8f __attribute__((ext_vector_type(8)));

#define N_NODES 65536
#define D_FEAT  64
#define N_EDGES 1048576

// Transposed-W LDS stride: 68 floats (even -> 8B-aligned b64 loads; bank
// pattern {4*lo, 4*lo+1} vs {4*lo+2, 4*lo+3} -> conflict-free across 64 banks).
#define WT_STRIDE 68

// ---------------------------------------------------------------------------
// Kernel 1: zero the aggregation scratch buffer (N_NODES * D_FEAT floats).
// ---------------------------------------------------------------------------
__global__ __launch_bounds__(256)
void gcnii_zero_kernel(float4* __restrict__ agg4) {
    int i = blockIdx.x * blockDim.x + threadIdx.x;
    agg4[i] = make_float4(0.f, 0.f, 0.f, 0.f);
}

// ---------------------------------------------------------------------------
// Kernel 2: edge-parallel SpMM scatter.
// 16 threads per edge; each thread handles 4 consecutive features (float4
// gather from x[src], native f32 atomics into agg[dst]).
// Inline asm guarantees GLOBAL_ATOMIC_ADD_F32 (no CAS-loop fallback); the
// non-returning form tracks on STOREcnt and is drained by s_endpgm's
// implicit wait-idle. WGP$ is read-only, so the RMW executes in L2's atomic
// units -> correct combining across all WGPs.
// ---------------------------------------------------------------------------
__global__ __launch_bounds__(256)
void gcnii_scatter_kernel(const float* __restrict__ x,
                          const int*   __restrict__ adj_src,
                          const int*   __restrict__ adj_dst,
                          const float* __restrict__ adj_val,
                          float*       __restrict__ agg) {
    const int tid  = blockIdx.x * blockDim.x + threadIdx.x;
    const int e    = tid >> 4;       // edge index
    const int part = tid & 15;       // which float4 of the 64-wide row

    const int   s = adj_src[e];
    const int   d = adj_dst[e];
    const float w = adj_val[e];

    const float4 xv = *(const float4*)(x + s * D_FEAT + part * 4);
    float* ap = agg + d * D_FEAT + part * 4;

    asm volatile(
        "global_atomic_add_f32 %0, %1, off\n\t"
        "global_atomic_add_f32 %0, %2, off offset:4\n\t"
        "global_atomic_add_f32 %0, %3, off offset:8\n\t"
        "global_atomic_add_f32 %0, %4, off offset:12"
        :
        : "v"(ap), "v"(w * xv.x), "v"(w * xv.y), "v"(w * xv.z), "v"(w * xv.w)
        : "memory");
}

// ---------------------------------------------------------------------------
// Kernel 3: irc = (1-alpha)*agg + alpha*feature ;
//           out = (1-beta)*irc + beta*(irc @ W)   via V_WMMA_F32_16X16X4_F32.
// Block = 256 threads = 8 waves. Each wave computes one 16-row tile across
// all 64 output columns (4 N-tiles), K accumulated in 16 steps of 4.
// A-fragment (f32 16x4): lane%16 = M row; VGPR v holds K = k0 + 2*(lane>>4) + v.
// B-fragment (f32 4x16): lane%16 = N col; VGPR v holds K = k0 + 2*(lane>>4) + v.
//   -> with W stored TRANSPOSED in LDS, each B fragment is one ds_load_b64.
// C/D layout (f32 16x16): VGPR v -> M = v + 8*(lane>>4), N = lane%16.
// ---------------------------------------------------------------------------
__global__ __launch_bounds__(256)
void gcnii_output_kernel(const float* __restrict__ agg,
                         const float* __restrict__ feature,
                         const float* __restrict__ weight,
                         const float* __restrict__ alphap,
                         const float* __restrict__ lamdap,
                         const int*   __restrict__ lp,
                         float*       __restrict__ out) {
    // sWt[c * WT_STRIDE + r] = W[r][c]  (transposed, padded)
    __shared__ float sWt[D_FEAT * WT_STRIDE];

    const int tid = threadIdx.x;

    // Cooperative transposed stage of W (64x64 f32): 256 threads x 4 float4s.
    {
#pragma unroll
        for (int i = 0; i < 4; ++i) {
            const int lin = i * 1024 + tid * 4;
            const int r   = lin >> 6;
            const int c   = lin & 63;
            const float4 wv = *(const float4*)(weight + lin);
            sWt[(c + 0) * WT_STRIDE + r] = wv.x;
            sWt[(c + 1) * WT_STRIDE + r] = wv.y;
            sWt[(c + 2) * WT_STRIDE + r] = wv.z;
            sWt[(c + 3) * WT_STRIDE + r] = wv.w;
        }
    }
    __syncthreads();

    const float alpha = alphap[0];
    const float beta  = __logf(lamdap[0] / (float)lp[0] + 1.0f);
    const float ia    = 1.0f - alpha;
    const float ib    = 1.0f - beta;

    const int wave = tid >> 5;
    const int lane = tid & 31;
    const int lo   = lane & 15;
    const int hi   = lane >> 4;

    const int m_base = (blockIdx.x * 8 + wave) * 16;
    const int row_a  = m_base + lo;

    v8f acc[4] = {};

    const float* aggA = agg     + row_a * D_FEAT;
    const float* feaA = feature + row_a * D_FEAT;

#pragma unroll
    for (int k0 = 0; k0 < D_FEAT; k0 += 4) {
        const int ka = k0 + hi * 2;            // this lane's K pair (even)
        const v2f ga = *(const v2f*)(aggA + ka);
        const v2f fa = *(const v2f*)(feaA + ka);
        v2f a;
        a.x = ia * ga.x + alpha * fa.x;        // irc in A-fragment layout
        a.y = ia * ga.y + alpha * fa.y;

        const float* bt = sWt + lo * WT_STRIDE + ka;
#pragma unroll
        for (int n = 0; n < 4; ++n) {
            // B fragment: two consecutive K values of column (n*16 + lo)
            const v2f b = *(const v2f*)(bt + n * 16 * WT_STRIDE);
            // 8 args: (neg_a, A, neg_b, B, c_mod, C, reuse_a, reuse_b)
            acc[n] = __builtin_amdgcn_wmma_f32_16x16x4_f32(
                false, a, false, b, (short)0, acc[n], false, false);
        }
    }

    // Epilogue in C/D layout: out = (1-beta)*irc + beta*(irc @ W)
#pragma unroll
    for (int n = 0; n < 4; ++n) {
        const int col = n * 16 + lo;
#pragma unroll
        for (int v = 0; v < 8; ++v) {
            const int row_c = m_base + v + hi * 8;
            const int idx   = row_c * D_FEAT + col;
            const float ircv = ia * agg[idx] + alpha * feature[idx];
            out[idx] = ib * ircv + beta * acc[n][v];
        }
    }
}

// ---------------------------------------------------------------------------
extern "C" void kernel_launch(void* const* d_in, const int* in_sizes, int n_in,
                              void* d_out, int out_size, void* d_ws, size_t ws_size,
                              hipStream_t stream) {
    const float* feature = (const float*)d_in[0];
    const float* x       = (const float*)d_in[1];
    const int*   adj_src = (const int*)  d_in[2];
    const int*   adj_dst = (const int*)  d_in[3];
    const float* adj_val = (const float*)d_in[4];
    const float* weight  = (const float*)d_in[5];
    const float* alphap  = (const float*)d_in[6];
    const float* lamdap  = (const float*)d_in[7];
    const int*   lp      = (const int*)  d_in[8];
    float*       out     = (float*)d_out;
    float*       agg     = (float*)d_ws;      // N_NODES * D_FEAT floats = 16 MB

    (void)in_sizes; (void)n_in; (void)out_size; (void)ws_size;

    // 1) agg = 0
    gcnii_zero_kernel<<<(N_NODES * D_FEAT / 4) / 256, 256, 0, stream>>>(
        (float4*)agg);

    // 2) agg[dst] += val * x[src]   (16 threads per edge, native f32 atomics)
    gcnii_scatter_kernel<<<(N_EDGES * 16) / 256, 256, 0, stream>>>(
        x, adj_src, adj_dst, adj_val, agg);

    // 3) residual mix + WMMA GEMM epilogue (8 waves/block, 128 rows/block)
    gcnii_output_kernel<<<N_NODES / 128, 256, 0, stream>>>(
        agg, feature, weight, alphap, lamdap, lp, out);
}